// GCNEncoder_446676599434
// MI455X (gfx1250) — compile-verified
//
#include <hip/hip_runtime.h>

#define NN 200000
#define NE 6400000
#define IN_DIM 32
#define HID_DIM 32
#define OUT_DIM 16

typedef __attribute__((ext_vector_type(2))) float v2f;
typedef __attribute__((ext_vector_type(8))) float v8f;

// ---------- degree / normalization ----------
__global__ void init_deg_kernel(float* __restrict__ deg, int n) {
    int i = blockIdx.x * blockDim.x + threadIdx.x;
    if (i < n) deg[i] = 1.0f;  // self-loop
}

__global__ void count_deg_kernel(const int* __restrict__ dst, float* __restrict__ deg, int e) {
    int i = blockIdx.x * blockDim.x + threadIdx.x;
    if (i < e) atomicAdd(&deg[dst[i]], 1.0f);
}

__global__ void rsqrt_kernel(float* __restrict__ deg, int n) {
    int i = blockIdx.x * blockDim.x + threadIdx.x;
    if (i < n) deg[i] = rsqrtf(deg[i]);
}

// ---------- GEMM with WMMA f32 16x16x4, epilogue scales row by dinv[row] ----------
// out[row, col] = (X @ W)[row, col] * dinv[row]
template <int K, int FOUT>
__global__ void gemm_scaled_kernel(const float* __restrict__ X,
                                   const float* __restrict__ W,
                                   const float* __restrict__ dinv,
                                   float* __restrict__ out, int nrows) {
    const int wave = (blockIdx.x * blockDim.x + threadIdx.x) >> 5;
    const int lane = threadIdx.x & 31;
    const int tilesN = FOUT / 16;
    const int tm = wave / tilesN;
    const int tn = wave % tilesN;
    if (tm * 16 >= nrows) return;  // wave-uniform: EXEC stays all-1s for WMMA

    const int m = lane & 15;      // row (A) / col (B,C,D) within tile
    const int half = lane >> 4;   // K-pair select (A,B) / row+8 select (C,D)
    const int row0 = tm * 16;
    const int col0 = tn * 16;

    v8f acc = {};
#pragma unroll
    for (int kb = 0; kb < K; kb += 4) {
        const int k0 = kb + 2 * half;
        v2f a, b;
        a.x = X[(size_t)(row0 + m) * K + k0 + 0];
        a.y = X[(size_t)(row0 + m) * K + k0 + 1];
        b.x = W[(size_t)(k0 + 0) * FOUT + col0 + m];
        b.y = W[(size_t)(k0 + 1) * FOUT + col0 + m];
        acc = __builtin_amdgcn_wmma_f32_16x16x4_f32(false, a, false, b,
                                                    (short)0, acc, false, false);
    }
#pragma unroll
    for (int r = 0; r < 8; ++r) {
        const int row = row0 + r + 8 * half;
        out[(size_t)row * FOUT + col0 + m] = acc[r] * dinv[row];
    }
}

// ---------- self-loop init + bias:  agg[i,f] = hws[i,f]*dinv[i] + bias[f] ----------
template <int F>
__global__ void init_agg_kernel(const float* __restrict__ hws,
                                const float* __restrict__ dinv,
                                const float* __restrict__ bias,
                                float* __restrict__ agg, int n) {
    int i = blockIdx.x * blockDim.x + threadIdx.x;
    if (i >= n * F) return;
    int node = i / F;
    int f = i - node * F;
    agg[i] = hws[i] * dinv[node] + bias[f];
}

// ---------- edge scatter:  agg[dst] += hws[src] * dinv[dst]  (F/4 lanes per edge) ----------
template <int F>
__global__ void edge_agg_kernel(const int* __restrict__ src, const int* __restrict__ dst,
                                const float* __restrict__ hws, const float* __restrict__ dinv,
                                float* __restrict__ agg, int ne) {
    constexpr int TPE = F / 4;  // float4 per lane
    int tid = blockIdx.x * blockDim.x + threadIdx.x;
    int e = tid / TPE;
    int g = tid - e * TPE;
    if (e >= ne) return;
    const int s = src[e];
    const int d = dst[e];
    const float nd = dinv[d];
    const float4 v = reinterpret_cast<const float4*>(hws + (size_t)s * F)[g];
    float* a = agg + (size_t)d * F + g * 4;
    atomicAdd(a + 0, v.x * nd);
    atomicAdd(a + 1, v.y * nd);
    atomicAdd(a + 2, v.z * nd);
    atomicAdd(a + 3, v.w * nd);
}

__global__ void relu_kernel(float* __restrict__ h, int n) {
    int i = blockIdx.x * blockDim.x + threadIdx.x;
    if (i < n) h[i] = fmaxf(h[i], 0.0f);
}

static inline int cdiv(long long a, long long b) { return (int)((a + b - 1) / b); }

extern "C" void kernel_launch(void* const* d_in, const int* in_sizes, int n_in,
                              void* d_out, int out_size, void* d_ws, size_t ws_size,
                              hipStream_t stream) {
    const float* x  = (const float*)d_in[0];          // [NN, 32]
    const int* ei   = (const int*)d_in[1];            // [2, NE]
    const float* W1 = (const float*)d_in[2];          // [32, 32]
    const float* b1 = (const float*)d_in[3];          // [32]
    const float* W2 = (const float*)d_in[4];          // [32, 16]
    const float* b2 = (const float*)d_in[5];          // [16]
    float* out = (float*)d_out;                       // [NN, 16]

    const int* src = ei;       // edge_index[0]
    const int* dst = ei + NE;  // edge_index[1]

    // workspace layout (all 256B-aligned by construction)
    char* ws = (char*)d_ws;
    float* dinv = (float*)ws;                                   // NN floats (0.8 MB)
    float* hw1  = (float*)(ws + (size_t)NN * 4);                // NN*32 (25.6 MB), dinv-scaled X@W1
    float* h1   = hw1 + (size_t)NN * HID_DIM;                   // NN*32 (25.6 MB), agg -> relu
    float* hw2  = h1  + (size_t)NN * HID_DIM;                   // NN*16 (12.8 MB), dinv-scaled h1@W2

    const int B = 256;

    // normalization
    init_deg_kernel<<<cdiv(NN, B), B, 0, stream>>>(dinv, NN);
    count_deg_kernel<<<cdiv(NE, B), B, 0, stream>>>(dst, dinv, NE);
    rsqrt_kernel<<<cdiv(NN, B), B, 0, stream>>>(dinv, NN);

    // layer 1: hw1 = (x @ W1) * dinv[row]
    {
        const int waves = (NN / 16) * (HID_DIM / 16);
        gemm_scaled_kernel<IN_DIM, HID_DIM>
            <<<cdiv((long long)waves * 32, B), B, 0, stream>>>(x, W1, dinv, hw1, NN);
    }
    init_agg_kernel<HID_DIM><<<cdiv((long long)NN * HID_DIM, B), B, 0, stream>>>(hw1, dinv, b1, h1, NN);
    edge_agg_kernel<HID_DIM><<<cdiv((long long)NE * (HID_DIM / 4), B), B, 0, stream>>>(
        src, dst, hw1, dinv, h1, NE);
    relu_kernel<<<cdiv((long long)NN * HID_DIM, B), B, 0, stream>>>(h1, NN * HID_DIM);

    // layer 2: hw2 = (h1 @ W2) * dinv[row]
    {
        const int waves = (NN / 16) * (OUT_DIM / 16);
        gemm_scaled_kernel<HID_DIM, OUT_DIM>
            <<<cdiv((long long)waves * 32, B), B, 0, stream>>>(h1, W2, dinv, hw2, NN);
    }
    init_agg_kernel<OUT_DIM><<<cdiv((long long)NN * OUT_DIM, B), B, 0, stream>>>(hw2, dinv, b2, out, NN);
    edge_agg_kernel<OUT_DIM><<<cdiv((long long)NE * (OUT_DIM / 4), B), B, 0, stream>>>(
        src, dst, hw2, dinv, out, NE);
}